// LorentzOscillatorLayer_71683004170582
// MI455X (gfx1250) — compile-verified
//
#include <hip/hip_runtime.h>
#include <hip/hip_bf16.h>

typedef __attribute__((ext_vector_type(2))) float v2f;
typedef __attribute__((ext_vector_type(8))) float v8f;

// ---------------------------------------------------------------------------
// Fold coupling matmuls into input transform (linearity):
//   Weff[d, h*3+k] = cs * sum_i Wi[d, i*3+k] * W_k[i, h]
// ---------------------------------------------------------------------------
__global__ void fuse_weights_kernel(const float* __restrict__ Wi,
                                    const float* __restrict__ Wx,
                                    const float* __restrict__ Wy,
                                    const float* __restrict__ Wz,
                                    const float* __restrict__ cs_p,
                                    float* __restrict__ Weff,
                                    int D, int H) {
  int idx = blockIdx.x * blockDim.x + threadIdx.x;
  int NH = 3 * H;
  if (idx >= D * NH) return;
  int d = idx / NH;
  int r = idx - d * NH;
  int h = r / 3;
  int k = r - h * 3;
  const float* Wk = (k == 0) ? Wx : (k == 1 ? Wy : Wz);
  const float* wrow = Wi + (size_t)d * NH + k;
  float acc = 0.0f;
  for (int i = 0; i < H; ++i)
    acc += wrow[i * 3] * Wk[(size_t)i * H + h];
  Weff[idx] = acc * (*cs_p);
}

// beff[h*3+k] = cs * sum_i bi[i*3+k] * W_k[i, h]
__global__ void fuse_bias_kernel(const float* __restrict__ bi,
                                 const float* __restrict__ Wx,
                                 const float* __restrict__ Wy,
                                 const float* __restrict__ Wz,
                                 const float* __restrict__ cs_p,
                                 float* __restrict__ beff,
                                 int H) {
  int idx = blockIdx.x * blockDim.x + threadIdx.x;
  int NH = 3 * H;
  if (idx >= NH) return;
  int h = idx / 3;
  int k = idx - h * 3;
  const float* Wk = (k == 0) ? Wx : (k == 1 ? Wy : Wz);
  float acc = 0.0f;
  for (int i = 0; i < H; ++i)
    acc += bi[i * 3 + k] * Wk[(size_t)i * H + h];
  beff[idx] = acc * (*cs_p);
}

// ---------------------------------------------------------------------------
// fp32 WMMA GEMM: C[M,N] = A[M,K] @ W[K,N] + bias[N]   (all row-major)
// One wave per 32(M) x 64(N) tile: per K-step, 2 A-fragment loads (b64) feed
// 8x V_WMMA_F32_16X16X4_F32 against 4 shared B fragments. All streaming
// pointers are rolling (strength-reduced), so the hot loop is loads + WMMA.
// Requires M%32==0, N%64==0, K%4==0.
// ---------------------------------------------------------------------------
__global__ void wmma_gemm_f32_kernel(const float* __restrict__ A,
                                     const float* __restrict__ W,
                                     const float* __restrict__ bias,
                                     float* __restrict__ C,
                                     int M, int N, int K) {
  const int wave   = blockIdx.x * (blockDim.x >> 5) + (threadIdx.x >> 5);
  const int mtiles = M >> 5;
  const int ntiles = N >> 6;
  if (wave >= mtiles * ntiles) return;          // wave-uniform: EXEC stays all-1s
  const int mt   = wave % mtiles;               // consecutive waves reuse same W slice
  const int ng   = wave / mtiles;
  const int lane = threadIdx.x & 31;
  const int half = lane >> 4;                   // 0: K pair {0,1}, 1: {2,3}
  const int l15  = lane & 15;
  const int m0   = mt << 5;
  const int n0   = ng << 6;

  v8f acc00 = {}, acc01 = {}, acc02 = {}, acc03 = {};
  v8f acc10 = {}, acc11 = {}, acc12 = {}, acc13 = {};

  // Rolling per-lane streaming pointers (advance by fixed strides only).
  const float* a0p = A + (size_t)(m0 + l15) * K + (half << 1);
  const float* a1p = a0p + (size_t)16 * K;
  const float* w0p = W + (size_t)(half << 1) * N + n0 + l15;  // row k+2h
  const float* w1p = w0p + N;                                 // row k+2h+1
  const size_t wstep = (size_t)4 * N;

  for (int k = 0; k < K; k += 4) {
    v2f a0 = *(const v2f*)a0p;                            // A[m   ][k+2h..k+2h+1]
    v2f a1 = *(const v2f*)a1p;                            // A[m+16][k+2h..k+2h+1]
    v2f b0 = { w0p[0],  w1p[0]  };                        // B[k+2h+v][n0+16j+l15]
    v2f b1 = { w0p[16], w1p[16] };
    v2f b2 = { w0p[32], w1p[32] };
    v2f b3 = { w0p[48], w1p[48] };
    a0p += 4;
    a1p += 4;
    w0p += wstep;
    w1p += wstep;
    acc00 = __builtin_amdgcn_wmma_f32_16x16x4_f32(false, a0, false, b0, (short)0, acc00, false, false);
    acc10 = __builtin_amdgcn_wmma_f32_16x16x4_f32(false, a1, false, b0, (short)0, acc10, false, false);
    acc01 = __builtin_amdgcn_wmma_f32_16x16x4_f32(false, a0, false, b1, (short)0, acc01, false, false);
    acc11 = __builtin_amdgcn_wmma_f32_16x16x4_f32(false, a1, false, b1, (short)0, acc11, false, false);
    acc02 = __builtin_amdgcn_wmma_f32_16x16x4_f32(false, a0, false, b2, (short)0, acc02, false, false);
    acc12 = __builtin_amdgcn_wmma_f32_16x16x4_f32(false, a1, false, b2, (short)0, acc12, false, false);
    acc03 = __builtin_amdgcn_wmma_f32_16x16x4_f32(false, a0, false, b3, (short)0, acc03, false, false);
    acc13 = __builtin_amdgcn_wmma_f32_16x16x4_f32(false, a1, false, b3, (short)0, acc13, false, false);
  }

  v8f accs[2][4] = {{acc00, acc01, acc02, acc03}, {acc10, acc11, acc12, acc13}};
  #pragma unroll
  for (int j = 0; j < 4; ++j) {
    const int col = n0 + (j << 4) + l15;
    const float bb = bias[col];
    #pragma unroll
    for (int mi = 0; mi < 2; ++mi) {
      #pragma unroll
      for (int v = 0; v < 8; ++v) {
        const int row = m0 + (mi << 4) + (half << 3) + v; // C row = 8*half + v
        C[(size_t)row * N + col] = accs[mi][j][v] + bb;
      }
    }
  }
}

// ---------------------------------------------------------------------------
// Sequential Lorenz RK4 scan. One thread per (b,h) oscillator; reads the
// coupling triple at (b,t,h) and overwrites it in place with the post-step
// state (exactly the A-matrix layout [B*S, 3H] col=h*3+k that GEMM2 needs).
// ---------------------------------------------------------------------------
__global__ void lorenz_scan_kernel(float* __restrict__ buf,          // [B*S, 3H]
                                   const float* __restrict__ prev,   // [B, H, 3]
                                   float* __restrict__ fstate,       // [B, H, 3]
                                   const float* __restrict__ sig_p,
                                   const float* __restrict__ rho_p,
                                   const float* __restrict__ beta_p,
                                   int B, int S, int H) {
  const int tid = blockIdx.x * blockDim.x + threadIdx.x;
  if (tid >= B * H) return;
  const float sigma = *sig_p, rho = *rho_p, beta = *beta_p;
  const float dt = 0.01f;
  const float sixth = 1.0f / 6.0f;
  const int b = tid / H;
  const int h = tid - b * H;
  const int NH = 3 * H;

  float sx = prev[tid * 3 + 0];
  float sy = prev[tid * 3 + 1];
  float sz = prev[tid * 3 + 2];

  float* p = buf + (size_t)b * S * NH + 3 * h;
  for (int t = 0; t < S; ++t, p += NH) {
    const float c0 = p[0], c1 = p[1], c2 = p[2];
    #pragma unroll
    for (int it = 0; it < 10; ++it) {
      // k1
      float dx = sigma * (sy - sx) + c0;
      float dy = sx * (rho - sz) - sy + c1;
      float dz = sx * sy - beta * sz + c2;
      float k1x = dt * dx, k1y = dt * dy, k1z = dt * dz;
      // k2
      float tx = sx + 0.5f * k1x, ty = sy + 0.5f * k1y, tz = sz + 0.5f * k1z;
      dx = sigma * (ty - tx) + c0;
      dy = tx * (rho - tz) - ty + c1;
      dz = tx * ty - beta * tz + c2;
      float k2x = dt * dx, k2y = dt * dy, k2z = dt * dz;
      // k3
      tx = sx + 0.5f * k2x; ty = sy + 0.5f * k2y; tz = sz + 0.5f * k2z;
      dx = sigma * (ty - tx) + c0;
      dy = tx * (rho - tz) - ty + c1;
      dz = tx * ty - beta * tz + c2;
      float k3x = dt * dx, k3y = dt * dy, k3z = dt * dz;
      // k4
      tx = sx + k3x; ty = sy + k3y; tz = sz + k3z;
      dx = sigma * (ty - tx) + c0;
      dy = tx * (rho - tz) - ty + c1;
      dz = tx * ty - beta * tz + c2;
      float k4x = dt * dx, k4y = dt * dy, k4z = dt * dz;

      sx += (k1x + 2.0f * (k2x + k3x) + k4x) * sixth;
      sy += (k1y + 2.0f * (k2y + k3y) + k4y) * sixth;
      sz += (k1z + 2.0f * (k2z + k3z) + k4z) * sixth;
    }
    p[0] = sx; p[1] = sy; p[2] = sz;   // state for output GEMM
  }
  fstate[tid * 3 + 0] = sx;
  fstate[tid * 3 + 1] = sy;
  fstate[tid * 3 + 2] = sz;
}

// ---------------------------------------------------------------------------
extern "C" void kernel_launch(void* const* d_in, const int* in_sizes, int n_in,
                              void* d_out, int out_size, void* d_ws, size_t ws_size,
                              hipStream_t stream) {
  const float* x    = (const float*)d_in[0];
  const float* prev = (const float*)d_in[1];
  const float* Wi   = (const float*)d_in[2];
  const float* bi   = (const float*)d_in[3];
  const float* Wo   = (const float*)d_in[4];
  const float* bo   = (const float*)d_in[5];
  const float* Wx   = (const float*)d_in[6];
  const float* Wy   = (const float*)d_in[7];
  const float* Wz   = (const float*)d_in[8];
  const float* sig  = (const float*)d_in[9];
  const float* rho  = (const float*)d_in[10];
  const float* beta = (const float*)d_in[11];
  const float* cs   = (const float*)d_in[12];

  const int B = 64, S = 512, D = 256, H = 256;
  const int NH = 3 * H;            // 768
  const int Mrows = B * S;         // 32768

  float* buf  = (float*)d_ws;                       // [Mrows, 3H]: coupling -> states (in place)
  float* Weff = buf + (size_t)Mrows * NH;           // [D, 3H]
  float* beff = Weff + (size_t)D * NH;              // [3H]

  float* outs   = (float*)d_out;                    // [B, S, H]
  float* fstate = outs + (size_t)Mrows * H;         // [B, H, 3]

  // 1) fold the three H x H coupling matmuls into the input transform
  {
    int n = D * NH;
    fuse_weights_kernel<<<(n + 255) / 256, 256, 0, stream>>>(Wi, Wx, Wy, Wz, cs, Weff, D, H);
    fuse_bias_kernel<<<(NH + 255) / 256, 256, 0, stream>>>(bi, Wx, Wy, Wz, cs, beff, H);
  }
  // 2) coupling = x @ Weff + beff      [32768 x 768], K = 256
  {
    int waves  = (Mrows / 32) * (NH / 64);
    int blocks = (waves * 32 + 255) / 256;
    wmma_gemm_f32_kernel<<<blocks, 256, 0, stream>>>(x, Weff, beff, buf, Mrows, NH, D);
  }
  // 3) sequential RK4 scan (in-place coupling -> per-timestep state)
  {
    int n = B * H;
    lorenz_scan_kernel<<<(n + 255) / 256, 256, 0, stream>>>(buf, prev, fstate, sig, rho, beta, B, S, H);
  }
  // 4) outs = states @ Wo + bo         [32768 x 256], K = 768
  {
    int waves  = (Mrows / 32) * (H / 64);
    int blocks = (waves * 32 + 255) / 256;
    wmma_gemm_f32_kernel<<<blocks, 256, 0, stream>>>(buf, Wo, bo, outs, Mrows, H, NH);
  }
}